// ClockworkDecoder_82695300317751
// MI455X (gfx1250) — compile-verified
//
#include <hip/hip_runtime.h>
#include <math.h>

#define T_STEPS 262144
#define HID 96
#define TILE 32                 // beliefs rows per async prefetch tile
#define W12_STRIDE 100          // padded LDS row stride (floats) for W_r rows 32..95

typedef float v2f __attribute__((ext_vector_type(2)));
typedef float v4f __attribute__((ext_vector_type(4)));
typedef float v8f __attribute__((ext_vector_type(8)));

#if __has_builtin(__builtin_amdgcn_tanhf)
#define FAST_TANH(x) __builtin_amdgcn_tanhf(x)   // single V_TANH_F32 on gfx1250
#else
#define FAST_TANH(x) tanhf(x)
#endif

// ---------------------------------------------------------------------------
// Kernel 1: serial clockwork scan. One wave32; lane i owns hidden row i of
// each group. Steady-state critical path: 1 LDS broadcast read (beliefs),
// 4+32 FMAs, v_tanh, LDS store + 8x ds_load_b128 h0 re-broadcast.
// Beliefs tiles stream in via global_load_async_to_lds_b128 (ASYNCcnt).
// ---------------------------------------------------------------------------
__global__ __launch_bounds__(32, 1)
void cw_scan_kernel(const float* __restrict__ beliefs,
                    const float* __restrict__ W_in,
                    const float* __restrict__ b_in,
                    const float* __restrict__ W_r,
                    float* __restrict__ H0,   // [T,32]
                    float* __restrict__ H1,   // [T/8,32]
                    float* __restrict__ H2)   // [T/64,32]
{
    __shared__ __align__(16) float s_bel[2][TILE * 4];
    __shared__ __align__(16) float s_h[HID];
    __shared__ __align__(16) float s_w12[64 * W12_STRIDE];

    const int lane = threadIdx.x;

    // per-lane constants
    const v4f win0 = *(const v4f*)(W_in + lane * 4);
    const v4f win1 = *(const v4f*)(W_in + (32 + lane) * 4);
    const v4f win2 = *(const v4f*)(W_in + (64 + lane) * 4);
    const float bi0 = b_in[lane], bi1 = b_in[32 + lane], bi2 = b_in[64 + lane];

    // row `lane` of W_r lives entirely in VGPRs (hot path: cols 0..31)
    float w0row[96];
#pragma unroll
    for (int q = 0; q < 24; ++q) {
        v4f w = *(const v4f*)(W_r + lane * 96 + q * 4);
        w0row[q * 4 + 0] = w.x; w0row[q * 4 + 1] = w.y;
        w0row[q * 4 + 2] = w.z; w0row[q * 4 + 3] = w.w;
    }

    // rows 32..95 of W_r into LDS (padded stride vs bank conflicts)
    for (int idx = lane; idx < 64 * 96; idx += 32) {
        int rr = idx / 96, cc = idx - rr * 96;
        s_w12[rr * W12_STRIDE + cc] = W_r[(32 + rr) * 96 + cc];
    }
    for (int j = lane; j < HID; j += 32) s_h[j] = 0.0f;

    float h0r[32];
#pragma unroll
    for (int j = 0; j < 32; ++j) h0r[j] = 0.0f;
    float cc1 = 0.0f, cc2 = 0.0f;   // cached W01*h1 and W02*h2

    __syncthreads();

    // prefetch tile 0: lane L copies beliefs row L (16B) into s_bel[0]
    {
        unsigned lds_off = (unsigned)(unsigned long long)(&s_bel[0][lane * 4]);
        unsigned long long ga =
            (unsigned long long)(beliefs + (size_t)lane * 4);
        asm volatile("global_load_async_to_lds_b128 %0, %1, off"
                     :: "v"(lds_off), "v"(ga) : "memory");
    }

    for (int t = 0; t < T_STEPS; ++t) {
        const int r = t & (TILE - 1);
        const int buf = (t / TILE) & 1;
        if (r == 0) {
            asm volatile("s_wait_asynccnt 0" ::: "memory");
            if (t + TILE < T_STEPS) {
                const int nb = buf ^ 1;
                unsigned lds_off =
                    (unsigned)(unsigned long long)(&s_bel[nb][lane * 4]);
                unsigned long long ga = (unsigned long long)(
                    beliefs + (size_t)(t + TILE + lane) * 4);
                asm volatile("global_load_async_to_lds_b128 %0, %1, off"
                             :: "v"(lds_off), "v"(ga) : "memory");
            }
        }
        const bool fire1 = (t & 7) == 0;
        const bool fire2 = (t & 63) == 0;

        const v4f bel = *(const v4f*)&s_bel[buf][r * 4];  // wave broadcast

        // ---- group-0 candidate (every step) ----
        float acc = bi0 + cc1 + cc2
                  + win0.x * bel.x + win0.y * bel.y
                  + win0.z * bel.z + win0.w * bel.w;
#pragma unroll
        for (int j = 0; j < 32; ++j) acc += w0row[j] * h0r[j];
        const float cand0 = FAST_TANH(acc);

        float cand1 = 0.0f, cand2 = 0.0f;
        if (fire1) {
            // replicate full (old) h: h0 from regs, h1/h2 via broadcast loads
            float hrep[96];
#pragma unroll
            for (int j = 0; j < 32; ++j) hrep[j] = h0r[j];
#pragma unroll
            for (int q = 0; q < 16; ++q) {
                v4f hv = *(const v4f*)&s_h[32 + q * 4];
                hrep[32 + q * 4 + 0] = hv.x; hrep[32 + q * 4 + 1] = hv.y;
                hrep[32 + q * 4 + 2] = hv.z; hrep[32 + q * 4 + 3] = hv.w;
            }
            float a1 = bi1 + win1.x * bel.x + win1.y * bel.y
                           + win1.z * bel.z + win1.w * bel.w;
#pragma unroll
            for (int q = 0; q < 24; ++q) {
                v4f w = *(const v4f*)&s_w12[lane * W12_STRIDE + q * 4];
                a1 += w.x * hrep[q * 4 + 0] + w.y * hrep[q * 4 + 1]
                    + w.z * hrep[q * 4 + 2] + w.w * hrep[q * 4 + 3];
            }
            cand1 = FAST_TANH(a1);
            if (fire2) {
                float a2 = bi2 + win2.x * bel.x + win2.y * bel.y
                               + win2.z * bel.z + win2.w * bel.w;
#pragma unroll
                for (int q = 0; q < 24; ++q) {
                    v4f w = *(const v4f*)&s_w12[(32 + lane) * W12_STRIDE + q * 4];
                    a2 += w.x * hrep[q * 4 + 0] + w.y * hrep[q * 4 + 1]
                        + w.z * hrep[q * 4 + 2] + w.w * hrep[q * 4 + 3];
                }
                cand2 = FAST_TANH(a2);
            }
        }

        // ---- commit (same-wave DS ops are in order; no barrier needed) ----
        s_h[lane] = cand0;
        H0[(size_t)t * 32 + lane] = cand0;
        if (fire1) { s_h[32 + lane] = cand1; H1[(size_t)(t >> 3) * 32 + lane] = cand1; }
        if (fire2) { s_h[64 + lane] = cand2; H2[(size_t)(t >> 6) * 32 + lane] = cand2; }

        // re-replicate h0 into registers (8x ds_load_b128, broadcast)
#pragma unroll
        for (int q = 0; q < 8; ++q) {
            v4f hv = *(const v4f*)&s_h[q * 4];
            h0r[q * 4 + 0] = hv.x; h0r[q * 4 + 1] = hv.y;
            h0r[q * 4 + 2] = hv.z; h0r[q * 4 + 3] = hv.w;
        }
        // refresh cached slow-group contributions with NEW h1/h2
        if (fire1) {
            float s = 0.0f;
#pragma unroll
            for (int q = 0; q < 8; ++q) {
                v4f hv = *(const v4f*)&s_h[32 + q * 4];
                s += w0row[32 + q * 4 + 0] * hv.x + w0row[32 + q * 4 + 1] * hv.y
                   + w0row[32 + q * 4 + 2] * hv.z + w0row[32 + q * 4 + 3] * hv.w;
            }
            cc1 = s;
        }
        if (fire2) {
            float s = 0.0f;
#pragma unroll
            for (int q = 0; q < 8; ++q) {
                v4f hv = *(const v4f*)&s_h[64 + q * 4];
                s += w0row[64 + q * 4 + 0] * hv.x + w0row[64 + q * 4 + 1] * hv.y
                   + w0row[64 + q * 4 + 2] * hv.z + w0row[64 + q * 4 + 3] * hv.w;
            }
            cc2 = s;
        }
    }
}

// ---------------------------------------------------------------------------
// Kernel 2: output heads via V_WMMA_F32_16X16X4_F32.
// One wave = 16 timesteps; D = A(16x4 slice of reconstructed H) x B(4x16,
// cols 0/1 = w_theta/w_z) accumulated over 24 K-chunks; bias preloaded in C.
// B is built branchlessly from an LDS select table (row 2 = zeros) so EXEC
// stays all-1s and no scalar branches sit between consecutive WMMAs.
// ---------------------------------------------------------------------------
__global__ __launch_bounds__(256, 2)
void cw_out_kernel(const float* __restrict__ H0,
                   const float* __restrict__ H1,
                   const float* __restrict__ H2,
                   const float* __restrict__ w_theta,
                   const float* __restrict__ b_theta,
                   const float* __restrict__ w_z,
                   const float* __restrict__ b_z,
                   float* __restrict__ theta_out,
                   float* __restrict__ z_out)
{
    __shared__ __align__(16) float s_w[3][HID];  // [0]=w_theta [1]=w_z [2]=0
    __shared__ float s_bias[4];                  // [0]=b_theta [1]=b_z [2,3]=0

    for (int i = threadIdx.x; i < HID; i += 256) {
        s_w[0][i] = w_theta[i];
        s_w[1][i] = w_z[i];
        s_w[2][i] = 0.0f;
    }
    if (threadIdx.x == 0) {
        s_bias[0] = b_theta[0]; s_bias[1] = b_z[0];
        s_bias[2] = 0.0f;       s_bias[3] = 0.0f;
    }
    __syncthreads();

    const int lane = threadIdx.x & 31;
    const int wave = threadIdx.x >> 5;
    const int t0   = (blockIdx.x * 8 + wave) * 16;
    const int nl   = lane & 15;     // N (B/C column), also A's M index
    const int half = lane >> 4;     // selects K sub-pair / M upper half
    const int kb   = half * 2;
    const int m    = nl;
    const int sel  = (nl < 2) ? nl : 2;   // branchless B row select

    v8f c;
    {
        const float bias = s_bias[sel];
#pragma unroll
        for (int i = 0; i < 8; ++i) c[i] = bias;
    }

#pragma unroll
    for (int ch = 0; ch < 24; ++ch) {
        const int K0 = ch * 4;
        const int k0 = K0 + kb;
        v2f a;
        if (K0 < 32) {
            const float* row = H0 + (size_t)(t0 + m) * 32;
            a.x = row[k0];      a.y = row[k0 + 1];
        } else if (K0 < 64) {
            const float* row = H1 + (size_t)((t0 + m) >> 3) * 32;
            a.x = row[k0 - 32]; a.y = row[k0 - 31];
        } else {
            const float* row = H2 + (size_t)((t0 + m) >> 6) * 32;
            a.x = row[k0 - 64]; a.y = row[k0 - 63];
        }
        const v2f b = *(const v2f*)&s_w[sel][k0];   // one ds_load_b64, no branch

        c = __builtin_amdgcn_wmma_f32_16x16x4_f32(false, a, false, b,
                                                  (short)0, c, false, false);
    }

#pragma unroll
    for (int rr = 0; rr < 8; ++rr) {
        const int tm = t0 + half * 8 + rr;
        if (nl == 0)      theta_out[tm] = c[rr];
        else if (nl == 1) z_out[tm]     = c[rr];
    }
}

// ---------------------------------------------------------------------------
extern "C" void kernel_launch(void* const* d_in, const int* in_sizes, int n_in,
                              void* d_out, int out_size, void* d_ws, size_t ws_size,
                              hipStream_t stream) {
    const float* beliefs = (const float*)d_in[0];
    const float* W_in    = (const float*)d_in[1];
    const float* b_in    = (const float*)d_in[2];
    const float* W_r     = (const float*)d_in[3];
    const float* w_theta = (const float*)d_in[4];
    const float* b_theta = (const float*)d_in[5];
    const float* w_z     = (const float*)d_in[6];
    const float* b_z     = (const float*)d_in[7];

    float* theta_out = (float*)d_out;
    float* z_out     = theta_out + T_STEPS;

    float* H0 = (float*)d_ws;                          // T   x 32
    float* H1 = H0 + (size_t)T_STEPS * 32;             // T/8 x 32
    float* H2 = H1 + (size_t)(T_STEPS / 8) * 32;       // T/64x 32

    cw_scan_kernel<<<1, 32, 0, stream>>>(beliefs, W_in, b_in, W_r, H0, H1, H2);
    cw_out_kernel<<<T_STEPS / 128, 256, 0, stream>>>(H0, H1, H2, w_theta,
                                                     b_theta, w_z, b_z,
                                                     theta_out, z_out);
}